// GCN_6047313953595
// MI455X (gfx1250) — compile-verified
//
#include <hip/hip_runtime.h>
#include <math.h>

#define NN 100000
#define EE 1600000
#define DD 64
#define LL 4
#define EPSV 1e-5f

typedef float v2f __attribute__((ext_vector_type(2)));
typedef float v8f __attribute__((ext_vector_type(8)));

// ---------------- small utility kernels ----------------

__global__ void zero_f32(float* __restrict__ p, int n) {
    int i = blockIdx.x * blockDim.x + threadIdx.x;
    if (i < n) p[i] = 0.0f;
}

__global__ void deg_count(const int* __restrict__ col, float* __restrict__ deg) {
    int e = blockIdx.x * blockDim.x + threadIdx.x;
    if (e < EE) atomicAdd(&deg[col[e]], 1.0f);
}

__global__ void make_dis(float* __restrict__ deg_dis) {
    int i = blockIdx.x * blockDim.x + threadIdx.x;
    if (i < NN) deg_dis[i] = rsqrtf(deg_dis[i] + 1.0f);
}

// ---------------- WMMA fp32 GEMM: H = X @ W  (N x 64  *  64 x 64) ----------------
// One wave computes a 16x64 output strip using V_WMMA_F32_16X16X4_F32.
// A layout (16x4 f32): lanes 0-15 hold row M=lane, K={k0,k0+1}; lanes 16-31 hold
// row M=lane-16, K={k0+2,k0+3}  -> contiguous float2 per lane.
// B layout (4x16 f32): VGPR0 = K=(k0|k0+2) across lanes, VGPR1 = K=(k0+1|k0+3).
// C layout (16x16 f32): VGPR r = row r (lanes 0-15) / row r+8 (lanes 16-31).
__global__ __launch_bounds__(256) void gemm_wmma(const float* __restrict__ X,
                                                 const float* __restrict__ W,
                                                 float* __restrict__ H) {
    int gid  = blockIdx.x * blockDim.x + threadIdx.x;
    int wave = gid >> 5;
    int lane = gid & 31;
    int m_base = wave * 16;            // wave-uniform
    if (m_base >= NN) return;          // uniform exit: EXEC all-ones inside

    int half = lane >> 4;              // 0 -> K pair base 0 ; 1 -> K pair base 2
    int lrow = lane & 15;

    v8f acc0 = {}; v8f acc1 = {}; v8f acc2 = {}; v8f acc3 = {};

    const float* xrow = X + (size_t)(m_base + lrow) * DD + 2 * half;

#pragma unroll
    for (int k0 = 0; k0 < DD; k0 += 4) {
        v2f a = *(const v2f*)(xrow + k0);          // 8B-aligned contiguous pair
        int kb = k0 + 2 * half;
        const float* wp = W + (size_t)kb * DD + lrow;
        v2f b0, b1, b2, b3;
        b0.x = wp[0];   b0.y = wp[DD];
        b1.x = wp[16];  b1.y = wp[DD + 16];
        b2.x = wp[32];  b2.y = wp[DD + 32];
        b3.x = wp[48];  b3.y = wp[DD + 48];
        acc0 = __builtin_amdgcn_wmma_f32_16x16x4_f32(false, a, false, b0, (short)0, acc0, false, false);
        acc1 = __builtin_amdgcn_wmma_f32_16x16x4_f32(false, a, false, b1, (short)0, acc1, false, false);
        acc2 = __builtin_amdgcn_wmma_f32_16x16x4_f32(false, a, false, b2, (short)0, acc2, false, false);
        acc3 = __builtin_amdgcn_wmma_f32_16x16x4_f32(false, a, false, b3, (short)0, acc3, false, false);
    }

    float* hout = H + (size_t)(m_base + 8 * half) * DD + lrow;
#pragma unroll
    for (int r = 0; r < 8; ++r) {
        hout[(size_t)r * DD +  0] = acc0[r];
        hout[(size_t)r * DD + 16] = acc1[r];
        hout[(size_t)r * DD + 32] = acc2[r];
        hout[(size_t)r * DD + 48] = acc3[r];
    }
}

// ---------------- agg = snorm * h + bias + x_res  (fused init) ----------------
__global__ void init_agg(const float* __restrict__ H, const float* __restrict__ Xin,
                         const float* __restrict__ dis, const float* __restrict__ bias,
                         float* __restrict__ agg) {
    int t = blockIdx.x * blockDim.x + threadIdx.x;
    if (t >= NN * 16) return;
    int i = t >> 4;
    int d = (t & 15) * 4;
    float s = dis[i]; s = s * s;               // snorm = 1/deg
    float4 hv = *(const float4*)(H   + (size_t)i * DD + d);
    float4 xv = *(const float4*)(Xin + (size_t)i * DD + d);
    float4 bv = *(const float4*)(bias + d);
    float4 o;
    o.x = s * hv.x + bv.x + xv.x;
    o.y = s * hv.y + bv.y + xv.y;
    o.z = s * hv.z + bv.z + xv.z;
    o.w = s * hv.w + bv.w + xv.w;
    *(float4*)(agg + (size_t)i * DD + d) = o;
}

// ---------------- edge scatter: agg[col] += h[row] * dis[row]*dis[col] ----------------
// One wave per edge; lane handles features (lane, lane+32) -> coalesced 128B atomics.
__global__ __launch_bounds__(256) void scatter_edges(const float* __restrict__ H,
                                                     const float* __restrict__ dis,
                                                     const int* __restrict__ ei,
                                                     float* __restrict__ agg) {
    int gid  = blockIdx.x * blockDim.x + threadIdx.x;
    int wave = gid >> 5;
    int lane = gid & 31;
    if (wave >= EE) return;
    int r = ei[wave];
    int c = ei[EE + wave];
    float coeff = dis[r] * dis[c];
    const float* hr = H + (size_t)r * DD;
    float* ac = agg + (size_t)c * DD;
    atomicAdd(ac + lane,      hr[lane]      * coeff);
    atomicAdd(ac + lane + 32, hr[lane + 32] * coeff);
}

// ---------------- per-node LayerNorm + exact GELU ----------------
__global__ __launch_bounds__(256) void ln_gelu(const float* __restrict__ agg,
                                               const float* __restrict__ gamma,
                                               const float* __restrict__ beta,
                                               float* __restrict__ out) {
    int gid  = blockIdx.x * blockDim.x + threadIdx.x;
    int node = gid >> 5;
    int lane = gid & 31;
    if (node >= NN) return;
    float2 v = *(const float2*)(agg + (size_t)node * DD + lane * 2);

    float sum = v.x + v.y;
#pragma unroll
    for (int off = 16; off > 0; off >>= 1) sum += __shfl_xor(sum, off, 32);
    float mean = sum * (1.0f / 64.0f);

    float dx = v.x - mean, dy = v.y - mean;
    float s2 = dx * dx + dy * dy;
#pragma unroll
    for (int off = 16; off > 0; off >>= 1) s2 += __shfl_xor(s2, off, 32);
    float inv = rsqrtf(s2 * (1.0f / 64.0f) + EPSV);

    float2 g  = *(const float2*)(gamma + lane * 2);
    float2 bb = *(const float2*)(beta  + lane * 2);
    float hx = dx * inv * g.x + bb.x;
    float hy = dy * inv * g.y + bb.y;

    const float kRSQRT2 = 0.70710678118654752f;
    float2 o;
    o.x = 0.5f * hx * (1.0f + erff(hx * kRSQRT2));
    o.y = 0.5f * hy * (1.0f + erff(hy * kRSQRT2));
    *(float2*)(out + (size_t)node * DD + lane * 2) = o;
}

// ---------------- host launcher ----------------
extern "C" void kernel_launch(void* const* d_in, const int* in_sizes, int n_in,
                              void* d_out, int out_size, void* d_ws, size_t ws_size,
                              hipStream_t stream) {
    const float* x      = (const float*)d_in[0];
    const int*   ei     = (const int*)  d_in[1];   // [2,E]: row = ei[0:E), col = ei[E:2E)
    const float* Ws     = (const float*)d_in[2];   // [L,64,64]
    const float* bs     = (const float*)d_in[3];   // [L,64]
    const float* gammas = (const float*)d_in[4];   // [L,64]
    const float* betas  = (const float*)d_in[5];   // [L,64]
    float* out = (float*)d_out;

    float* ws  = (float*)d_ws;
    float* dis = ws;                               // N
    float* H   = ws + NN;                          // N*64
    float* AGG = H  + (size_t)NN * DD;             // N*64
    float* X0  = AGG + (size_t)NN * DD;            // N*64
    float* X1  = X0  + (size_t)NN * DD;            // N*64

    // degree -> dis = rsqrt(deg + 1)
    zero_f32 <<<(NN + 255) / 256, 256, 0, stream>>>(dis, NN);
    deg_count<<<(EE + 255) / 256, 256, 0, stream>>>(ei + EE, dis);
    make_dis <<<(NN + 255) / 256, 256, 0, stream>>>(dis);

    const int gemm_blocks    = ((NN / 16) * 32 + 255) / 256;  // 6250 waves
    const int init_blocks    = (NN * 16 + 255) / 256;
    const int scatter_blocks = (int)(((long long)EE * 32 + 255) / 256);
    const int ln_blocks      = (NN * 32 + 255) / 256;

    const float* xin = x;
    for (int l = 0; l < LL; ++l) {
        float* xout = (l == LL - 1) ? out : ((l & 1) ? X1 : X0);
        gemm_wmma    <<<gemm_blocks,    256, 0, stream>>>(xin, Ws + (size_t)l * DD * DD, H);
        init_agg     <<<init_blocks,    256, 0, stream>>>(H, xin, dis, bs + l * DD, AGG);
        scatter_edges<<<scatter_blocks, 256, 0, stream>>>(H, dis, ei, AGG);
        ln_gelu      <<<ln_blocks,      256, 0, stream>>>(AGG, gammas + l * DD, betas + l * DD, xout);
        xin = xout;
    }
}